// PositionalEncoding_54915451847173
// MI455X (gfx1250) — compile-verified
//
#include <hip/hip_runtime.h>
#include <stdint.h>

// PositionalEncoding: out[b, s, d] = (d even) ? sin(s / 10000^(d/D)) : cos(...)
// Pure store-bandwidth problem (1 GiB out). Compute each unique (sin,cos) once
// into LDS, then broadcast to all 32 batch slots with CDNA5 async
// LDS->global stores (ASYNCcnt), waiting once at the end.

#define SEQ          4096
#define DMODEL       2048
#define BATCH        32
#define HALF_D       (DMODEL / 2)          // 1024
#define SD           (SEQ * DMODEL)        // 8,388,608 floats
#define SD_BYTES     ((uint64_t)SD * 4ull) // 32 MiB per batch
#define THREADS      256
#define CHUNK_FLOATS 4096                  // per block (16 KiB LDS)
#define CHUNK_PAIRS  (CHUNK_FLOATS / 2)    // 2048 (sin,cos) pairs
#define CHUNK_BYTES  (CHUNK_FLOATS * 4)    // 16384
#define PAIRS_PER_T  (CHUNK_PAIRS / THREADS) // 8
#define SWEEPS       (CHUNK_BYTES / (THREADS * 16)) // 4 x b128 per lane per batch

typedef int v4i __attribute__((ext_vector_type(4)));
typedef __attribute__((address_space(1))) v4i gm_v4i_t;   // global b128
typedef __attribute__((address_space(3))) v4i lds_v4i_t;  // LDS b128

#ifdef __has_builtin
#if __has_builtin(__builtin_amdgcn_global_store_async_from_lds_b128)
#define HAVE_ASYNC_STORE_BUILTIN 1
#endif
#if __has_builtin(__builtin_amdgcn_s_wait_asynccnt)
#define HAVE_WAIT_ASYNC_BUILTIN 1
#endif
#endif

__device__ __forceinline__ void async_store_lds_b128(uint64_t gaddr, uint32_t lds_off) {
#if defined(HAVE_ASYNC_STORE_BUILTIN)
    __builtin_amdgcn_global_store_async_from_lds_b128(
        (gm_v4i_t*)gaddr, (lds_v4i_t*)lds_off, 0, 0);
#else
    // VGLOBAL async encoding: VADDR = 64-bit global address pair,
    // VSRC = LDS byte address, saddr off. Tracked with ASYNCcnt.
    asm volatile("global_store_async_from_lds_b128 %0, %1, off"
                 :: "v"(gaddr), "v"(lds_off)
                 : "memory");
#endif
}

__device__ __forceinline__ void wait_asynccnt0() {
#if defined(HAVE_WAIT_ASYNC_BUILTIN)
    __builtin_amdgcn_s_wait_asynccnt(0);
#else
    asm volatile("s_wait_asynccnt 0" ::: "memory");
#endif
}

__global__ __launch_bounds__(THREADS)
void PositionalEncoding_54915451847173_kernel(float* __restrict__ out) {
    __shared__ float2 lds[CHUNK_PAIRS];   // 16 KiB, 8-byte aligned

    const uint32_t tid      = threadIdx.x;
    const uint32_t blk      = blockIdx.x;
    const uint32_t basePair = blk * CHUNK_PAIRS;

    // theta = s * 10000^{-i/1024} = s * exp2(i * (-log2(10000)/1024))
    const float NEG_L2F   = -0.012976281620653759f;   // -log2(10000)/1024
    const float INV_2PI   =  0.15915494309189535f;
    const float TWO_PI    =  6.283185307179586f;

#pragma unroll
    for (int k = 0; k < PAIRS_PER_T; ++k) {
        const uint32_t lp = (uint32_t)k * THREADS + tid;   // local pair idx
        const uint32_t p  = basePair + lp;                 // global pair idx
        const uint32_t s  = p >> 10;                       // p / HALF_D
        const uint32_t i  = p & (HALF_D - 1);

        const float invf  = exp2f((float)i * NEG_L2F);     // v_exp_f32
        const float theta = (float)s * invf;

        // explicit mod-2pi reduction so fast hardware sin/cos stay in range
        float r = theta * INV_2PI;
        r -= rintf(r);                                     // v_rndne_f32
        const float rad = r * TWO_PI;

        lds[lp] = make_float2(__sinf(rad), __cosf(rad));   // ds_store_b64
    }

    __syncthreads();  // LDS writes visible before async engine reads them

    const uint32_t lds_base = (uint32_t)(uintptr_t)(&lds[0]);
    const uint64_t gbase    = (uint64_t)(uintptr_t)out
                            + (uint64_t)blk * (uint64_t)CHUNK_BYTES;

    for (int b = 0; b < BATCH; ++b) {
        const uint64_t gb = gbase + (uint64_t)b * SD_BYTES;
#pragma unroll
        for (int k = 0; k < SWEEPS; ++k) {
            const uint32_t off = (uint32_t)k * (THREADS * 16) + tid * 16;
            async_store_lds_b128(gb + off, lds_base + off);
        }
    }

    // Drain async stores before LDS dealloc / wave exit.
    wait_asynccnt0();
}

extern "C" void kernel_launch(void* const* d_in, const int* in_sizes, int n_in,
                              void* d_out, int out_size, void* d_ws, size_t ws_size,
                              hipStream_t stream) {
    (void)d_in; (void)in_sizes; (void)n_in; (void)d_ws; (void)ws_size; (void)out_size;
    float* out = (float*)d_out;
    const int grid = SD / CHUNK_FLOATS;   // 2048 blocks
    PositionalEncoding_54915451847173_kernel<<<grid, THREADS, 0, stream>>>(out);
}